// YOLOLayer_6055903887553
// MI455X (gfx1250) — compile-verified
//
#include <hip/hip_runtime.h>
#include <stdint.h>

#define NAK 4
#define NTK 64
#define NGH 76
#define NGW 136
#define NCELLS (NGH * NGW)   // 10336
#define NB 16
#define PLANE (NAK * NCELLS) // 41344

typedef __attribute__((ext_vector_type(2))) float   v2f;
typedef __attribute__((ext_vector_type(8))) float   v8f;
typedef __attribute__((ext_vector_type(4))) unsigned int vu4;
typedef __attribute__((ext_vector_type(8))) int     vi8;
typedef __attribute__((ext_vector_type(4))) int     vi4;

// Python scalars may arrive as int32 or float32 single-element arrays;
// disambiguate by bit pattern (1088 as float bits is a huge int).
__device__ __forceinline__ float scalar_to_float(const void* p) {
    int iv = *(const int*)p;
    if (iv > 0 && iv < (1 << 20)) return (float)iv;
    return *(const float*)p;
}

__global__ __launch_bounds__(256) void yolo_assign_kernel(
    const float* __restrict__ targets,  // (NB, NTK, 6): cls, tid, x, y, w, h
    const float* __restrict__ anchors,  // (NAK, 2) in pixels
    const void*  __restrict__ img_w_p,
    float* __restrict__ out)            // [tconf | tbox | tid] flat f32
{
    __shared__ float  raw[NTK * 6];     // staged raw targets for this batch
    __shared__ float4 sMin[NTK];        // gxmin, gymin, gxmax, gymax
    __shared__ float4 sBox[NTK];        // gx, gy, gw, gh
    __shared__ float  sGA[NTK];         // gw * gh
    __shared__ float  sId[NTK];         // track id (as float)
    __shared__ float4 sU0[NTK];         // u0[t][a] = pa[a] + ga[t] + eps

    const int   tid   = threadIdx.x;
    const int   b     = blockIdx.y;
    const float img_w = scalar_to_float(img_w_p);
    const float stride = img_w / (float)NGW;          // 8.0
    const float inv_stride = 1.0f / stride;

    // ---------------- stage targets for batch b into LDS --------------------
    const float* tsrc = targets + (size_t)b * (NTK * 6);
#if __has_builtin(__builtin_amdgcn_tensor_load_to_lds) && __has_builtin(__builtin_amdgcn_s_wait_tensorcnt)
    if (tid < 32) {  // one wave issues the TDM DMA; EXEC-independent, wave-level
        uint32_t lds_off = (uint32_t)(uintptr_t)(&raw[0]);
        uint64_t gaddr   = (uint64_t)(uintptr_t)tsrc;
        vu4 g0;
        g0.x = 1u;                                        // count = 1 descriptor
        g0.y = lds_off;                                   // LDS byte address
        g0.z = (uint32_t)(gaddr & 0xFFFFFFFFull);         // global addr [31:0]
        g0.w = (uint32_t)((gaddr >> 32) & 0x01FFFFFFull)  // global addr [56:32]
             | (2u << 30);                                // type = 2 ("image")
        vi8 g1;
        g1[0] = (int)(2u << 16);      // data_size = 4 bytes
        g1[1] = (int)(384u << 16);    // tensor_dim0 = 384 (lo16 in [31:16])
        g1[2] = (int)(1u << 16);      // tensor_dim0 hi = 0 | tensor_dim1 = 1
        g1[3] = (int)(384u << 16);    // tensor_dim1 hi = 0 | tile_dim0 = 384
        g1[4] = 1;                    // tile_dim1 = 1, tile_dim2 = 0
        g1[5] = 384;                  // tensor_dim0_stride lo
        g1[6] = (int)(384u << 16);    // stride0 hi = 0 | tensor_dim1_stride lo
        g1[7] = 0;                    // tensor_dim1_stride hi
        vi4 gz4 = {0, 0, 0, 0};                 // groups 2/3 unused (<=2D tensor)
        vi8 gz8 = {0, 0, 0, 0, 0, 0, 0, 0};     // extra group (clang-23 6-arg form)
        __builtin_amdgcn_tensor_load_to_lds(g0, g1, gz4, gz4, gz8, 0);
        __builtin_amdgcn_s_wait_tensorcnt(0);
    }
#else
    for (int i = tid; i < NTK * 6; i += 256) raw[i] = tsrc[i];
#endif
    __syncthreads();

    // ---------------- per-target GT box params ------------------------------
    if (tid < NTK) {
        float gx = raw[tid * 6 + 2] * (float)NGW;
        float gy = raw[tid * 6 + 3] * (float)NGH;
        gx = fminf(fmaxf(gx, 0.0f), (float)NGW - 1.0f);
        gy = fminf(fmaxf(gy, 0.0f), (float)NGH - 1.0f);
        float gw = raw[tid * 6 + 4] * (float)NGW;
        float gh = raw[tid * 6 + 5] * (float)NGH;
        sMin[tid] = make_float4(gx - 0.5f * gw, gy - 0.5f * gh,
                                gx + 0.5f * gw, gy + 0.5f * gh);
        sBox[tid] = make_float4(gx, gy, gw, gh);
        sGA[tid]  = gw * gh;
        sId[tid]  = raw[tid * 6 + 1];
    }
    __syncthreads();

    // ---------------- anchor constants --------------------------------------
    float aw[NAK], ah[NAK];
#pragma unroll
    for (int a = 0; a < NAK; ++a) {
        aw[a] = anchors[2 * a + 0] * inv_stride;
        ah[a] = anchors[2 * a + 1] * inv_stride;
    }

    // ---------------- WMMA: union base table u0[a][t] = pa[a]+ga[t]+eps -----
    // D = A x B + C with 16x16x4 f32 WMMA (wave 0 only, EXEC all ones).
    //   row r = a*4 + g (g = target group t>>4), col c = t & 15
    //   A[r][k] = (r&3 == k) ? 1 : 0   (one-hot group selector)
    //   B[k][c] = ga[k*16 + c]
    //   C[r][c] = pa[r>>2] + 1e-16
    if (tid < 32) {
        float pa[NAK];
#pragma unroll
        for (int a = 0; a < NAK; ++a) pa[a] = aw[a] * ah[a];

        const int L  = tid & 31;
        const int hi = L >> 4;   // 0/1 half-wave
        const int lo = L & 15;
        // A (16x4 f32): lane holds M = lo, K = hi*2 + j (VGPR j)
        v2f A, B;
        A.x = ((lo & 3) == (hi * 2 + 0)) ? 1.0f : 0.0f;
        A.y = ((lo & 3) == (hi * 2 + 1)) ? 1.0f : 0.0f;
        // B (4x16 f32): lane holds K = hi*2 + j, N = lo
        B.x = sGA[(hi * 2 + 0) * 16 + lo];
        B.y = sGA[(hi * 2 + 1) * 16 + lo];
        v8f C;
#pragma unroll
        for (int v = 0; v < 8; ++v)
            C[v] = pa[(v >> 2) + hi * 2] + 1e-16f;

        v8f D = __builtin_amdgcn_wmma_f32_16x16x4_f32(
            /*neg_a=*/false, A, /*neg_b=*/false, B,
            /*c_mod=*/(short)0, C, /*reuse_a=*/false, /*reuse_b=*/false);

        float* pU0 = (float*)sU0;
#pragma unroll
        for (int v = 0; v < 8; ++v) {
            int r = v + hi * 8;            // C/D layout: VGPR v -> M = v + 8*hi
            int a = r >> 2;
            int t = (r & 3) * 16 + lo;
            pU0[t * 4 + a] = D[v];
        }
    }
    __syncthreads();

    // ---------------- main: one cell per thread, all 4 anchors --------------
    const int c = blockIdx.x * 256 + tid;
    if (c >= NCELLS) return;
    const float px = (float)(c % NGW);
    const float py = (float)(c / NGW);

    float pxmin[NAK], pxmax[NAK], pymin[NAK], pymax[NAK];
    float inv_pw[NAK], inv_ph[NAK];
    float bi[NAK], bu[NAK];
    int   bidx[NAK];
#pragma unroll
    for (int a = 0; a < NAK; ++a) {
        pxmin[a] = px - 0.5f * aw[a];  pxmax[a] = px + 0.5f * aw[a];
        pymin[a] = py - 0.5f * ah[a];  pymax[a] = py + 0.5f * ah[a];
        inv_pw[a] = 1.0f / aw[a];      inv_ph[a] = 1.0f / ah[a];
        bi[a] = 0.0f; bu[a] = 1.0f; bidx[a] = 0;
    }

    // argmax over IoU via cross-multiplication (unions are strictly positive);
    // strict '>' keeps the first max, matching jnp.argmax tie-breaking.
    for (int t = 0; t < NTK; ++t) {
        float4 m  = sMin[t];
        float4 u0 = sU0[t];
        float u0a[4] = {u0.x, u0.y, u0.z, u0.w};
#pragma unroll
        for (int a = 0; a < NAK; ++a) {
            float ix = fminf(pxmax[a], m.z) - fmaxf(pxmin[a], m.x);
            float iy = fminf(pymax[a], m.w) - fmaxf(pymin[a], m.y);
            ix = fmaxf(ix, 0.0f);
            iy = fmaxf(iy, 0.0f);
            float inter = ix * iy;
            float u     = u0a[a] - inter;           // pa + ga + eps - inter
            bool  upd   = inter * bu[a] > bi[a] * u;
            bi[a]   = upd ? inter : bi[a];
            bu[a]   = upd ? u     : bu[a];
            bidx[a] = upd ? t     : bidx[a];
        }
    }

    // ---------------- epilogue: tconf, tbox, tid -----------------------------
    float* outConf = out;
    float* outBox  = out + (size_t)NB * PLANE;                     // 661504
    float* outId   = out + (size_t)NB * PLANE * 5;                 // 3307520
#pragma unroll
    for (int a = 0; a < NAK; ++a) {
        float iou = bi[a] / bu[a];
        bool  fg  = iou > 0.5f;
        bool  ign = (iou < 0.5f) && (iou > 0.4f);
        float conf = fg ? 1.0f : (ign ? -1.0f : 0.0f);
        float4 g   = sBox[bidx[a]];
        float  idv = fg ? sId[bidx[a]] : -1.0f;
        float4 tb;
        tb.x = fg ? (g.x - px) * inv_pw[a]      : 0.0f;
        tb.y = fg ? (g.y - py) * inv_ph[a]      : 0.0f;
        tb.z = fg ? __logf(g.z * inv_pw[a])     : 0.0f;
        tb.w = fg ? __logf(g.w * inv_ph[a])     : 0.0f;
        size_t idx = (size_t)(b * NAK + a) * NCELLS + c;
        outConf[idx] = conf;
        ((float4*)outBox)[idx] = tb;
        outId[idx] = idv;
    }
}

extern "C" void kernel_launch(void* const* d_in, const int* in_sizes, int n_in,
                              void* d_out, int out_size, void* d_ws, size_t ws_size,
                              hipStream_t stream) {
    // setup_inputs order: p_cat (unused data), targets, anchors, img_w, img_h
    const float* targets = (const float*)d_in[1];
    const float* anchors = (const float*)d_in[2];
    const void*  img_w   = d_in[3];
    float* out = (float*)d_out;
    (void)in_sizes; (void)n_in; (void)out_size; (void)d_ws; (void)ws_size;

    dim3 grid((NCELLS + 255) / 256, NB, 1);
    yolo_assign_kernel<<<grid, 256, 0, stream>>>(targets, anchors, img_w, out);
}